// CMMLunit_50173807952434
// MI455X (gfx1250) — compile-verified
//
#include <hip/hip_runtime.h>

// CDNA5 (gfx1250) wave32 WMMA types
typedef float v2f __attribute__((ext_vector_type(2)));
typedef float v8f __attribute__((ext_vector_type(8)));

// D = A(16x4) * B(4x16) + C, f32. 8-arg VOP3P form:
// (neg_a, A, neg_b, B, c_mod, C, reuse_a, reuse_b)
#define WMMA_F32(a, b, c) \
  __builtin_amdgcn_wmma_f32_16x16x4_f32(false, (a), false, (b), (short)0, (c), false, false)

constexpr int B_ = 8, N_ = 4096, D_ = 128, H_ = 512;

// Issue an async global->LDS copy of 32 bytes (two b128) per lane.
// gfx1250 GLOBAL_LOAD_ASYNC_TO_LDS: copies memory at VADDR+off into LDS at VDST+off, tracked by ASYNCcnt.
__device__ __forceinline__ void async_copy_32B(unsigned lds_addr,
                                               unsigned long long gaddr) {
  asm volatile(
      "global_load_async_to_lds_b128 %0, %1, off\n\t"
      "global_load_async_to_lds_b128 %0, %1, off offset:16"
      :
      : "v"(lds_addr), "v"(gaddr)
      : "memory");
}
__device__ __forceinline__ void wait_async0() {
  asm volatile("s_wait_asynccnt 0x0" ::: "memory");
}
__device__ __forceinline__ void wait_ds0() {
  asm volatile("s_wait_dscnt 0x0" ::: "memory");
}

// ---------------------------------------------------------------------------
// Kernel 1: sq[row] = ||r_row||^2   (B*N rows of 128 f32)
// ---------------------------------------------------------------------------
__global__ __launch_bounds__(256) void sq_kernel(const float* __restrict__ r,
                                                 float* __restrict__ sq) {
  int row = blockIdx.x * 256 + threadIdx.x;  // 32768 rows total
  const float4* p = (const float4*)(r + (size_t)row * D_);
  float s = 0.f;
#pragma unroll
  for (int i = 0; i < D_ / 4; ++i) {
    float4 v = p[i];
    s += v.x * v.x + v.y * v.y + v.z * v.z + v.w * v.w;
  }
  sq[row] = s;
}

// ---------------------------------------------------------------------------
// Kernel 2: fused RBF attention (flash-style, f32 WMMA).
//   S = r_i . r_j^T (K=128); w = exp(-max(sq_i + sq_j - 2S, 0))
//   acc += w @ r_col ; rowsum += w @ ones ; r2 = acc/rowsum + r_i
// Block = 256 threads = 8 waves = 128 rows. Column tiles double-buffered in
// LDS and filled with GLOBAL_LOAD_ASYNC_TO_LDS_B128 (ASYNCcnt), one barrier
// per column step.
// ---------------------------------------------------------------------------
__global__ __launch_bounds__(256) void rbf_attn_kernel(const float* __restrict__ r,
                                                       const float* __restrict__ sq,
                                                       float* __restrict__ r2) {
  __shared__ float scol[2][16 * 128];   // double-buffered 16x128 column tiles
  __shared__ float wst[8][16 * 16];     // per-wave w tile staging (D- -> A-layout)

  const int lane = threadIdx.x & 31;
  const int wv   = threadIdx.x >> 5;
  const int l15  = lane & 15;
  const int h    = lane >> 4;           // lane half (ISA fragment layouts)
  const int b    = blockIdx.y;
  const int i0   = blockIdx.x * 128 + wv * 16;

  const float* rb  = r  + (size_t)b * N_ * D_;
  const float* sqb = sq + (size_t)b * N_;

  // Cooperative tile footprint for this thread (32B of a 16x128 tile)
  const int trow = threadIdx.x >> 4;
  const int tcol = (threadIdx.x & 15) * 8;
  const unsigned lds0 = (unsigned)(size_t)(&scol[0][trow * 128 + tcol]);
  const unsigned lds1 = (unsigned)(size_t)(&scol[1][trow * 128 + tcol]);
  const float* gtile = rb + (size_t)trow * D_ + tcol;  // j0 = 0 base

  // Preload A fragments for this wave's 16x128 row tile:
  // afrag[k] = r[i0+l15][4k + 2h + {0,1}]
  v2f afrag[32];
  {
    const float* arow = rb + (size_t)(i0 + l15) * D_ + 2 * h;
#pragma unroll
    for (int k = 0; k < 32; ++k) afrag[k] = *(const v2f*)(arow + 4 * k);
  }
  float sqA[8];
#pragma unroll
  for (int g = 0; g < 8; ++g) sqA[g] = sqb[i0 + g + 8 * h];

  v8f acc[8];
  v8f rs = {};
#pragma unroll
  for (int t = 0; t < 8; ++t) acc[t] = v8f{};
  v2f ones; ones.x = 1.f; ones.y = 1.f;

  // Prologue: kick off async copy of tile 0 into buffer 0
  async_copy_32B(lds0, (unsigned long long)(size_t)gtile);

  for (int jt = 0; jt < N_ / 16; ++jt) {
    const int cur = jt & 1;
    const float* sc = &scol[cur][0];

    wait_async0();     // my async stores to LDS have landed
    __syncthreads();   // tile[cur] ready for all; tile[cur^1] free to overwrite

    // Overlap: issue async copy of the next column tile into the other buffer
    if (jt + 1 < N_ / 16) {
      async_copy_32B(cur ? lds0 : lds1,
                     (unsigned long long)(size_t)(gtile + (size_t)(jt + 1) * 16 * D_));
    }
    float sqj = sqb[jt * 16 + l15];

    // S tile: 16x16, K=128 -> 32 chained f32 WMMAs.
    // B-frag = r_col[l15][4k + 2h + {0,1}]
    v8f c = {};
#pragma unroll
    for (int k = 0; k < 32; ++k) {
      v2f bfrag = *(const v2f*)(sc + l15 * 128 + 4 * k + 2 * h);
      c = WMMA_F32(afrag[k], bfrag, c);
    }

    // w = exp(-max(d2,0)); stage into per-wave LDS region in [m][k] order
#pragma unroll
    for (int g = 0; g < 8; ++g) {
      float d2 = sqA[g] + sqj - 2.f * c[g];
      d2 = fmaxf(d2, 0.f);
      wst[wv][(g + 8 * h) * 16 + l15] = __expf(-d2);
    }
    wait_ds0();  // wave-private staging: DS ops are in-order per wave

    v2f wa[4];
#pragma unroll
    for (int kk = 0; kk < 4; ++kk)
      wa[kk] = *(const v2f*)(&wst[wv][l15 * 16 + 4 * kk + 2 * h]);

    // acc += w @ r_col (K=16, N=128 in 8 tiles); rowsum += w @ ones
#pragma unroll
    for (int kk = 0; kk < 4; ++kk) {
      rs = WMMA_F32(wa[kk], ones, rs);
#pragma unroll
      for (int t = 0; t < 8; ++t) {
        v2f bfrag;
        bfrag.x = sc[(4 * kk + 2 * h) * 128 + l15 + 16 * t];
        bfrag.y = sc[(4 * kk + 2 * h + 1) * 128 + l15 + 16 * t];
        acc[t] = WMMA_F32(wa[kk], bfrag, acc[t]);
      }
    }
  }

  // epilogue: r2 = acc / rowsum + r  (i.e. (w + I) @ r)
#pragma unroll
  for (int g = 0; g < 8; ++g) {
    float inv = 1.0f / rs[g];
    size_t rowoff = (size_t)(b * N_ + i0 + g + 8 * h) * D_;
    const float* rrow = rb + (size_t)(i0 + g + 8 * h) * D_;
#pragma unroll
    for (int t = 0; t < 8; ++t) {
      int col = l15 + 16 * t;
      r2[rowoff + col] = acc[t][g] * inv + rrow[col];
    }
  }
}

// ---------------------------------------------------------------------------
// Kernel 3: FFN  out = leaky_relu(r2 @ W1 + b1) @ W2 + b2
// Per wave: 16-row tile; loop H in chunks of 16 (W1/W2 chunks staged in LDS).
// ---------------------------------------------------------------------------
__global__ __launch_bounds__(256) void ffn_kernel(const float* __restrict__ r2,
                                                  const float* __restrict__ W1,
                                                  const float* __restrict__ b1,
                                                  const float* __restrict__ W2,
                                                  const float* __restrict__ b2,
                                                  float* __restrict__ out) {
  __shared__ float w1c[128 * 16];   // W1[:, chunk]
  __shared__ float w2c[16 * 128];   // W2[chunk, :]
  __shared__ float hst[8][16 * 16]; // per-wave h tile staging

  const int lane = threadIdx.x & 31;
  const int wv   = threadIdx.x >> 5;
  const int l15  = lane & 15;
  const int h    = lane >> 4;
  const int row0 = blockIdx.x * 128 + wv * 16;

  v2f afrag[32];
  {
    const float* arow = r2 + (size_t)(row0 + l15) * D_ + 2 * h;
#pragma unroll
    for (int k = 0; k < 32; ++k) afrag[k] = *(const v2f*)(arow + 4 * k);
  }
  float b2v[8];
#pragma unroll
  for (int t = 0; t < 8; ++t) b2v[t] = b2[l15 + 16 * t];

  v8f acc[8];
#pragma unroll
  for (int t = 0; t < 8; ++t) acc[t] = v8f{};

  for (int hc = 0; hc < H_ / 16; ++hc) {
    __syncthreads();
    {
      // stage W1 chunk [128][16]
      int k  = threadIdx.x >> 1;
      int c0 = (threadIdx.x & 1) * 8;
      const float4* s1 = (const float4*)(W1 + (size_t)k * H_ + hc * 16 + c0);
      float4* d1 = (float4*)(&w1c[k * 16 + c0]);
      d1[0] = s1[0];
      d1[1] = s1[1];
      // stage W2 chunk [16][128]
      int kr = threadIdx.x >> 4;
      int c1 = (threadIdx.x & 15) * 8;
      const float4* s2 = (const float4*)(W2 + (size_t)(hc * 16 + kr) * D_ + c1);
      float4* d2p = (float4*)(&w2c[kr * 128 + c1]);
      d2p[0] = s2[0];
      d2p[1] = s2[1];
    }
    float b1v = b1[hc * 16 + l15];
    __syncthreads();

    // h tile = r2_tile @ W1chunk  (16x16, K=128)
    v8f hcc = {};
#pragma unroll
    for (int k = 0; k < 32; ++k) {
      v2f bfrag;
      bfrag.x = w1c[(4 * k + 2 * h) * 16 + l15];
      bfrag.y = w1c[(4 * k + 2 * h + 1) * 16 + l15];
      hcc = WMMA_F32(afrag[k], bfrag, hcc);
    }
    // bias + leaky relu (slope 0.01 > 0 => leaky(x) == max(x, 0.01x)), stage
#pragma unroll
    for (int g = 0; g < 8; ++g) {
      float v = hcc[g] + b1v;
      hst[wv][(g + 8 * h) * 16 + l15] = fmaxf(v, 0.01f * v);
    }
    wait_ds0();  // wave-private staging

    v2f ha[4];
#pragma unroll
    for (int kk = 0; kk < 4; ++kk)
      ha[kk] = *(const v2f*)(&hst[wv][l15 * 16 + 4 * kk + 2 * h]);

    // out acc += h_chunk @ W2chunk (K=16, N=128)
#pragma unroll
    for (int kk = 0; kk < 4; ++kk) {
#pragma unroll
      for (int t = 0; t < 8; ++t) {
        v2f bfrag;
        bfrag.x = w2c[(4 * kk + 2 * h) * 128 + l15 + 16 * t];
        bfrag.y = w2c[(4 * kk + 2 * h + 1) * 128 + l15 + 16 * t];
        acc[t] = WMMA_F32(ha[kk], bfrag, acc[t]);
      }
    }
  }

#pragma unroll
  for (int g = 0; g < 8; ++g) {
    size_t rowoff = (size_t)(row0 + g + 8 * h) * D_;
#pragma unroll
    for (int t = 0; t < 8; ++t)
      out[rowoff + l15 + 16 * t] = acc[t][g] + b2v[t];
  }
}

// ---------------------------------------------------------------------------
extern "C" void kernel_launch(void* const* d_in, const int* in_sizes, int n_in,
                              void* d_out, int out_size, void* d_ws, size_t ws_size,
                              hipStream_t stream) {
  (void)in_sizes; (void)n_in; (void)out_size; (void)ws_size;
  const float* r  = (const float*)d_in[0];
  const float* W1 = (const float*)d_in[1];
  const float* b1 = (const float*)d_in[2];
  const float* W2 = (const float*)d_in[3];
  const float* b2 = (const float*)d_in[4];
  float* out = (float*)d_out;

  // workspace: sq (B*N f32 = 128KB) then r2 (B*N*D f32 = 16.8MB)
  float* sq = (float*)d_ws;
  float* r2 = sq + (size_t)B_ * N_;

  sq_kernel<<<dim3((B_ * N_) / 256), 256, 0, stream>>>(r, sq);
  rbf_attn_kernel<<<dim3(N_ / 128, B_), 256, 0, stream>>>(r, sq, r2);
  ffn_kernel<<<dim3((B_ * N_) / 128), 256, 0, stream>>>(r2, W1, b1, W2, b2, out);
}